// BBoxCDLoss_75076028334308
// MI455X (gfx1250) — compile-verified
//
#include <hip/hip_runtime.h>

typedef float v2f __attribute__((ext_vector_type(2)));
typedef float v8f __attribute__((ext_vector_type(8)));

// Decode bbox (center, size, euler ZXY) -> corner c (0..7): p = center + R*(signs*half)
__device__ __forceinline__ void bbox_corner(const float* __restrict__ b, int c,
                                            float& x, float& y, float& z) {
  float cx = b[0], cy = b[1], cz = b[2];
  float hx = b[3] * 0.5f, hy = b[4] * 0.5f, hz = b[5] * 0.5f;
  float az = b[6], ax = b[7], ay = b[8];
  float px = (c & 4) ? -hx : hx;
  float py = (c & 2) ? -hy : hy;
  float pz = (c & 1) ? -hz : hz;
  float sa = __sinf(az), ca = __cosf(az);   // Z
  float sb = __sinf(ax), cb = __cosf(ax);   // X
  float sc = __sinf(ay), cc = __cosf(ay);   // Y
  // R = Rz(a) * Rx(b) * Ry(c)
  float r00 = ca * cc - sa * sb * sc, r01 = -sa * cb, r02 = ca * sc + sa * sb * cc;
  float r10 = sa * cc + ca * sb * sc, r11 =  ca * cb, r12 = sa * sc - ca * sb * cc;
  float r20 = -cb * sc,               r21 =  sb,      r22 = cb * cc;
  x = cx + r00 * px + r01 * py + r02 * pz;
  y = cy + r10 * px + r11 * py + r12 * pz;
  z = cz + r20 * px + r21 * py + r22 * pz;
}

// One wave handles 2 boxes per iteration with one V_WMMA_F32_16X16X4_F32.
// Rows (M)  = tgt corners (the min axis) -> min is IN-LANE over D[0..7].
// Cols (N)  = src corners.
// A[m] = (tx,ty,tz,|t|^2), B[:,n] = (-2sx,-2sy,-2sz,1)
//   => D[m][n] = |t_m|^2 - 2 t_m . s_n ;  dist = D + |s_n|^2 (const over min axis)
__global__ void bbox_cd_partial(const float* __restrict__ src,
                                const float* __restrict__ tgt,
                                float* __restrict__ partial,
                                int nPairs) {
  const int lane = threadIdx.x & 31;
  const int wib  = threadIdx.x >> 5;
  const int wavesPerBlock = blockDim.x >> 5;
  const int waveId = blockIdx.x * wavesPerBlock + wib;
  const int nWaves = gridDim.x * wavesPerBlock;
  const int sub  = lane & 15;   // row (A) / col (B) index 0..15
  const int half = lane >> 4;   // 0: K=0,1 ; 1: K=2,3
  // Valid (box-matched) D columns: lanes 0-7 (box0) and 24-31 (box1).
  const bool validLane = (lane < 8) || (lane >= 24);
  float acc = 0.0f;

  for (int p = waveId; p < nPairs; p += nWaves) {
    const int box = p * 2 + (sub >> 3);   // rows/cols 0-7 -> box0, 8-15 -> box1
    const int c   = lane & 7;             // corner index
    float sx, sy, sz, tx, ty, tz;
    bbox_corner(src + (size_t)box * 9, c, sx, sy, sz);
    bbox_corner(tgt + (size_t)box * 9, c, tx, ty, tz);
    float ss = sx * sx + sy * sy + sz * sz;
    float tt = tx * tx + ty * ty + tz * tz;

    // A: 16x4 (MxK). lanes 0-15: K=0,1 ; lanes 16-31: K=2,3
    // B: 4x16 (KxN). same K split across lane halves.
    v2f A, B;
    A.x = half ? tz            : tx;
    A.y = half ? tt            : ty;
    B.x = half ? (-2.0f * sz)  : (-2.0f * sx);
    B.y = half ? 1.0f          : (-2.0f * sy);
    v8f Cz = {};
    v8f D = __builtin_amdgcn_wmma_f32_16x16x4_f32(
        /*neg_a=*/false, A, /*neg_b=*/false, B,
        /*c_mod=*/(short)0, Cz, /*reuse_a=*/false, /*reuse_b=*/false);

    // In-lane min over the 8 tgt-corner rows.
    float m01 = fminf(D[0], D[1]);
    float m23 = fminf(D[2], D[3]);
    float m45 = fminf(D[4], D[5]);
    float m67 = fminf(D[6], D[7]);
    float m   = fminf(fminf(m01, m23), fminf(m45, m67));

    acc += validLane ? (m + ss) : 0.0f;   // one src-corner row-min per valid lane
  }

  // wave32 reduction (once per wave)
  acc += __shfl_xor(acc, 16, 32);
  acc += __shfl_xor(acc,  8, 32);
  acc += __shfl_xor(acc,  4, 32);
  acc += __shfl_xor(acc,  2, 32);
  acc += __shfl_xor(acc,  1, 32);

  __shared__ float wsum[32];
  if (lane == 0) wsum[wib] = acc;
  __syncthreads();
  if (threadIdx.x == 0) {
    float b = 0.0f;
    for (int i = 0; i < wavesPerBlock; ++i) b += wsum[i];
    partial[blockIdx.x] = b;
  }
}

__global__ void bbox_cd_reduce(const float* __restrict__ partial, int n,
                               float* __restrict__ out, float scale) {
  __shared__ float sm[256];
  float s = 0.0f;
  for (int i = threadIdx.x; i < n; i += blockDim.x) s += partial[i];
  sm[threadIdx.x] = s;
  __syncthreads();
  for (int off = 128; off > 0; off >>= 1) {
    if ((int)threadIdx.x < off) sm[threadIdx.x] += sm[threadIdx.x + off];
    __syncthreads();
  }
  if (threadIdx.x == 0) out[0] = sm[0] * scale;
}

extern "C" void kernel_launch(void* const* d_in, const int* in_sizes, int n_in,
                              void* d_out, int out_size, void* d_ws, size_t ws_size,
                              hipStream_t stream) {
  const float* src = (const float*)d_in[0];
  const float* tgt = (const float*)d_in[1];
  float* out = (float*)d_out;
  float* ws  = (float*)d_ws;

  const int N = in_sizes[0] / 9;       // 262144 boxes
  const int nPairs = N / 2;            // 2 boxes per WMMA
  const int blocks = 256;              // 256 blocks x 8 waves = 2048 waves
  const int threads = 256;

  bbox_cd_partial<<<blocks, threads, 0, stream>>>(src, tgt, ws, nPairs);
  const float scale = 1.0f / (8.0f * (float)N);   // mean over N*8 row-minima
  bbox_cd_reduce<<<1, 256, 0, stream>>>(ws, blocks, out, scale);
}